// GAT_39427799777563
// MI455X (gfx1250) — compile-verified
//
#include <hip/hip_runtime.h>
#include <hip/hip_bf16.h>

// GAT forward_relation fused kernel for MI455X (gfx1250, wave32).
//
// Roofline: traffic ~1.06 GB (entity_embs + w_r dominate) vs ~1.3 GFLOP
// -> bandwidth bound, floor ~45us @ 23.3 TB/s. Strategy: single pass over
// the big [N,K,D] tensors with b128 loads, keep We in LDS (read-once from
// HBM, reused twice), fp32 throughout, and fuse the tiny 100x100 output
// projection as V_WMMA_F32_16X16X4_F32 tiles.

typedef float v2f __attribute__((ext_vector_type(2)));
typedef float v8f __attribute__((ext_vector_type(8)));

#define NROWS     40000
#define KN        32
#define DD        100
#define ROW_ELEMS (KN * DD)      // 3200 floats per row
#define TILE_M    16             // rows per block
#define NEG_INF_F (-9e15f)
#define ALPHA_F   0.2f

__global__ __launch_bounds__(256)
void GAT_39427799777563_kernel(const float* __restrict__ item,
                               const float* __restrict__ ent,
                               const float* __restrict__ wr,
                               const int*   __restrict__ adj,
                               const float* __restrict__ Wout,
                               const float* __restrict__ bout,
                               float*       __restrict__ out) {
    __shared__ float sW[112 * DD];     // W_out, rows padded/zeroed to 112
    __shared__ float sWe[ROW_ELEMS];   // We for current row
    __shared__ float sH[TILE_M * DD];  // h_prime tile
    __shared__ float sEsum[KN];
    __shared__ float sAtt[KN];

    const int tid  = threadIdx.x;
    const int n0   = blockIdx.x * TILE_M;

    // ---- stage W_out into LDS once per block (zero-pad rows 100..111) ----
    for (int idx = tid; idx < 112 * DD; idx += 256) {
        int row = idx / DD;
        int col = idx - row * DD;
        sW[idx] = (row < DD) ? Wout[row * DD + col] : 0.0f;
    }
    __syncthreads();

    // ---- per-row attention pipeline ----
    for (int r = 0; r < TILE_M; ++r) {
        const int n = n0 + r;
        const float* eP = ent + (size_t)n * ROW_ELEMS;
        const float* wP = wr  + (size_t)n * ROW_ELEMS;

        // prefetch next row's tensors (global_prefetch_b8)
        if (n + 1 < NROWS && tid * 13 < ROW_ELEMS) {
            __builtin_prefetch(ent + (size_t)(n + 1) * ROW_ELEMS + tid * 13, 0, 1);
            __builtin_prefetch(wr  + (size_t)(n + 1) * ROW_ELEMS + tid * 13, 0, 1);
        }

        // We = entity * w_r  (float4 => global_load_b128, coalesced)
        for (int i4 = tid; i4 < ROW_ELEMS / 4; i4 += 256) {
            float4 a = ((const float4*)eP)[i4];
            float4 b = ((const float4*)wP)[i4];
            float4 p;
            p.x = a.x * b.x; p.y = a.y * b.y; p.z = a.z * b.z; p.w = a.w * b.w;
            ((float4*)sWe)[i4] = p;
        }
        __syncthreads();

        // per-k row sums: 8 threads per k, shuffle-reduce within groups of 8
        {
            int k    = tid >> 3;   // 0..31
            int part = tid & 7;
            float s = 0.0f;
            for (int d = part; d < DD; d += 8) s += sWe[k * DD + d];
            s += __shfl_xor(s, 1, 32);
            s += __shfl_xor(s, 2, 32);
            s += __shfl_xor(s, 4, 32);
            if (part == 0) sEsum[k] = s;
        }
        __syncthreads();

        // masked softmax over K=32: exactly one wave32 (lane == k)
        if (tid < 32) {
            float s = sEsum[tid];
            float e = (s > 0.0f) ? s : ALPHA_F * s;          // leaky_relu
            float lg = (adj[(size_t)n * KN + tid] > 0) ? e : NEG_INF_F;
            float mx = lg;
            for (int m = 16; m >= 1; m >>= 1) mx = fmaxf(mx, __shfl_xor(mx, m, 32));
            float p = expf(lg - mx);
            float sum = p;
            for (int m = 16; m >= 1; m >>= 1) sum += __shfl_xor(sum, m, 32);
            sAtt[tid] = p / sum;
        }
        __syncthreads();

        // h_prime[r][d] = sum_k att[k] * We[k][d]   (lane d: conflict-free)
        if (tid < DD) {
            float acc = 0.0f;
            for (int k = 0; k < KN; ++k) acc += sAtt[k] * sWe[k * DD + tid];
            sH[r * DD + tid] = acc;
        }
        __syncthreads();
    }

    // ---- output projection: x = h_prime @ W_out^T + b_out + item ----
    // 7 waves, each owns a 16x16 (M x Ncol) tile; K = D = 100 in steps of 4.
    const int wv   = tid >> 5;
    const int lane = tid & 31;
    if (wv < 7) {
        const int j0 = wv * 16;
        const int mA = lane & 15;           // A: M index
        const int jn = lane & 15;           // B: N index
        const int kk = (lane >> 4) << 1;    // lane-half K offset

        v8f c = {};
        for (int k0 = 0; k0 < DD; k0 += 4) {
            v2f a, b;
            a.x = sH[mA * DD + k0 + kk];
            a.y = sH[mA * DD + k0 + kk + 1];
            b.x = sW[(j0 + jn) * DD + k0 + kk];
            b.y = sW[(j0 + jn) * DD + k0 + kk + 1];
            // V_WMMA_F32_16X16X4_F32: fp32 in/out, no precision loss
            c = __builtin_amdgcn_wmma_f32_16x16x4_f32(
                    false, a, false, b, (short)0, c, false, false);
        }

        const int ncol  = j0 + (lane & 15);
        const int mbase = (lane >> 4) * 8;  // C/D layout: M = v + 8*(lane/16)
        if (ncol < DD) {
            float bj = bout[ncol];
            for (int v = 0; v < 8; ++v) {
                size_t o = (size_t)(n0 + mbase + v) * DD + ncol;
                out[o] = c[v] + bj + item[o];
            }
        }
    }
}

extern "C" void kernel_launch(void* const* d_in, const int* in_sizes, int n_in,
                              void* d_out, int out_size, void* d_ws, size_t ws_size,
                              hipStream_t stream) {
    const float* item = (const float*)d_in[0];   // [N,100]
    const float* ent  = (const float*)d_in[1];   // [N,32,100]
    const float* wrp  = (const float*)d_in[2];   // [N,32,100]
    const int*   adj  = (const int*)d_in[3];     // [N,32]
    const float* Wout = (const float*)d_in[4];   // [100,100]
    const float* bout = (const float*)d_in[5];   // [100]
    float* out = (float*)d_out;                  // [N,100]

    dim3 grid(NROWS / TILE_M);   // 2500 blocks
    dim3 block(256);             // 8 wave32s
    GAT_39427799777563_kernel<<<grid, block, 0, stream>>>(
        item, ent, wrp, adj, Wout, bout, out);
}